// DynamicBaseLine_80865644249734
// MI455X (gfx1250) — compile-verified
//
#include <hip/hip_runtime.h>
#include <math.h>

// Causal sliding-window min, W = 128, via van Herk–Gil–Werman block scans.
// Bandwidth-bound: 64 MB total traffic -> ~2.9 us floor at 23.3 TB/s.
// CDNA5 paths used: global_load_async_to_lds_b128 (ASYNCcnt), s_wait_asynccnt,
// wave32 shuffle scans, b128 LDS/global accesses.

#define WG   256
#define W    128
#define TILE 2048                 // elements per workgroup (16 blocks of 128)
#define NBLK (TILE / W)           // 16 blocks
#define NWAVE (WG / 32)           // 8 waves

#if defined(__gfx1250__) && __has_builtin(__builtin_amdgcn_global_load_async_to_lds_b128)
#define HAVE_ASYNC_LDS 1
typedef int v4i __attribute__((ext_vector_type(4)));
typedef __attribute__((address_space(1))) v4i gv4i;   // global (device) int4
typedef __attribute__((address_space(3))) v4i lv4i;   // LDS int4
static __device__ __forceinline__ void async_copy16(const float* g, float* l) {
    // (global src, lds dst, imm offset, imm cpol)
    __builtin_amdgcn_global_load_async_to_lds_b128((gv4i*)g, (lv4i*)l, 0, 0);
}
static __device__ __forceinline__ void wait_async0() {
#if __has_builtin(__builtin_amdgcn_s_wait_asynccnt)
    __builtin_amdgcn_s_wait_asynccnt(0);
#else
    asm volatile("s_wait_asynccnt 0" ::: "memory");
#endif
}
#endif

__global__ __launch_bounds__(WG)
void slidemin128_kernel(const float* __restrict__ x, float* __restrict__ out, int n) {
    __shared__ float lds[TILE + W];   // [0,128) = left halo, [128, 128+TILE) = tile

    const int tid  = threadIdx.x;
    const int lane = tid & 31;
    const int wave = tid >> 5;
    const int tileStart = (int)blockIdx.x * TILE;
    const float INF = __builtin_inff();

    // ---- Stage tile + halo into LDS ----
#if defined(HAVE_ASYNC_LDS)
    {
        // body: 2 iterations x 256 lanes x 16B = 2048 floats
        #pragma unroll
        for (int it = 0; it < TILE / (WG * 4); ++it) {
            int g = tileStart + it * WG * 4 + tid * 4;
            if (g > n - 4) g = n - 4;          // safety clamp (no-op when TILE | n)
            async_copy16(x + g, &lds[W + it * WG * 4 + tid * 4]);
        }
        // halo: first wave copies the 128 elements preceding the tile
        if (tid < 32) {
            int g = tileStart - W + tid * 4;
            if (g < 0) g = 0;                  // clamped; replaced by +inf in regs
            async_copy16(x + g, &lds[tid * 4]);
        }
        wait_async0();
    }
#else
    {
        #pragma unroll
        for (int it = 0; it < TILE / (WG * 4); ++it) {
            int g = tileStart + it * WG * 4 + tid * 4;
            if (g > n - 4) g = n - 4;
            *(float4*)&lds[W + it * WG * 4 + tid * 4] = *(const float4*)(x + g);
        }
        if (tid < 32) {
            int g = tileStart - W + tid * 4;
            if (g < 0) g = 0;
            *(float4*)&lds[tid * 4] = *(const float4*)(x + g);
        }
    }
#endif
    __syncthreads();

    // ---- Per wave: process blocks wave, wave+8 (2 blocks each) ----
    for (int b = wave; b < NBLK; b += NWAVE) {
        // halo block (previous 128 elements) and current block, 4 floats per lane
        float4 y  = *(const float4*)&lds[b * W + lane * 4];        // halo
        float4 xc = *(const float4*)&lds[W + b * W + lane * 4];    // current

        // left edge of the whole sequence: halo is +inf
        if (blockIdx.x == 0 && b == 0) { y.x = INF; y.y = INF; y.z = INF; y.w = INF; }

        // --- halo block: suffix mins ---
        float s3 = y.w;
        float s2 = fminf(y.z, s3);
        float s1 = fminf(y.y, s2);
        float s0 = fminf(y.x, s1);
        float mH = s0;
        // reverse exclusive min-scan over lanes: f[l] = min(mH[l+1..31]), f[31]=INF
        float f = __shfl_down(mH, 1, 32);
        if (lane == 31) f = INF;
        #pragma unroll
        for (int d = 1; d < 32; d <<= 1) {
            float t = __shfl_down(f, d, 32);
            if (lane + d < 32) f = fminf(f, t);
        }
        float sufFull = fminf(s0, f);                 // suffix starting at offset lane*4
        float q4 = __shfl_down(sufFull, 1, 32);       // suffix at offset (lane+1)*4
        float q1 = fminf(s1, f);
        float q2 = fminf(s2, f);
        float q3 = fminf(s3, f);

        // --- current block: prefix mins ---
        float p0 = xc.x;
        float p1 = fminf(p0, xc.y);
        float p2 = fminf(p1, xc.z);
        float p3 = fminf(p2, xc.w);
        float mC = p3;
        // forward exclusive min-scan over lanes: e[l] = min(mC[0..l-1]), e[0]=INF
        float e = __shfl_up(mC, 1, 32);
        if (lane == 0) e = INF;
        #pragma unroll
        for (int d = 1; d < 32; d <<= 1) {
            float t = __shfl_up(e, d, 32);
            if (lane >= d) e = fminf(e, t);
        }

        // --- combine: out[o] = min(prefixC[o], suffixHalo[o+1]); o==127 -> prefix only
        float4 o4;
        o4.x = fminf(fminf(e, p0), q1);
        o4.y = fminf(fminf(e, p1), q2);
        o4.z = fminf(fminf(e, p2), q3);
        float P3 = fminf(e, p3);
        o4.w = (lane == 31) ? P3 : fminf(P3, q4);

        int gi = tileStart + b * W + lane * 4;
        if (gi + 4 <= n) *(float4*)(out + gi) = o4;
    }
}

extern "C" void kernel_launch(void* const* d_in, const int* in_sizes, int n_in,
                              void* d_out, int out_size, void* d_ws, size_t ws_size,
                              hipStream_t stream) {
    const float* x = (const float*)d_in[0];
    float* out = (float*)d_out;
    int n = in_sizes[0];
    int grid = (n + TILE - 1) / TILE;
    slidemin128_kernel<<<dim3(grid), dim3(WG), 0, stream>>>(x, out, n);
}